// MultiViewAttentionDNNBasedRegressor_55353538511276
// MI455X (gfx1250) — compile-verified
//
#include <hip/hip_runtime.h>
#include <hip/hip_bf16.h>
#include <math.h>

// ---------------------------------------------------------------------------
// MI455X (gfx1250) implementation.
// Problem: G=3000 genes, T=256 timepoints, H=128 hidden. f32 throughout.
// Dominant work: hg = relu(z @ W2 + b2) over all genes: ~25 GFLOP f32 ->
// V_WMMA_F32_16X16X4_F32, single pass with flash-softmax over genes.
// ---------------------------------------------------------------------------

typedef __attribute__((ext_vector_type(2))) float v2f;
typedef __attribute__((ext_vector_type(8))) float v8f;

#define Gn 3000
#define Tn 256
#define Hn 128
#define GPG 12      // genes per workgroup chunk
#define NCHUNK 250  // GPG * NCHUNK == Gn
#define H0_LD 132   // padded LDS stride (bank-conflict free across lane halves)

__device__ __forceinline__ v8f wmma_f32_16x16x4(v2f a, v2f b, v8f c) {
  // D = A(16x4,f32) x B(4x16,f32) + C(16x16,f32); 8-arg VOP3P form.
  return __builtin_amdgcn_wmma_f32_16x16x4_f32(
      /*neg_a=*/false, a, /*neg_b=*/false, b,
      /*c_mod=*/(short)0, c, /*reuse_a=*/false, /*reuse_b=*/false);
}

// ---------------------------------------------------------------------------
// Kernel A:  u = relu(x^T @ W1 + b1)   [T,H], K = G = 3000 (750 wmma steps)
// grid.x = 16 row tiles (T/16); 8 waves = 8 col tiles (H/16)
// ---------------------------------------------------------------------------
__global__ void __launch_bounds__(256) k_h_stage1(
    const float* __restrict__ x, const float* __restrict__ W1,
    const float* __restrict__ b1, float* __restrict__ u) {
  const int wave = threadIdx.x >> 5, lane = threadIdx.x & 31;
  const int lg = lane & 15, hi = lane >> 4;
  const int tm = blockIdx.x, tn = wave;
  const int t = tm * 16 + lg, n = tn * 16 + lg;
  v8f c = {};
  for (int kk = 0; kk < Gn / 4; ++kk) {
    const int k0 = (kk << 2) + (hi << 1);
    v2f a, b;
    a.x = x[k0 * Tn + t];           // A[m=t][k=g] = x[g][t] (transposed read)
    a.y = x[(k0 + 1) * Tn + t];
    b.x = W1[k0 * Hn + n];
    b.y = W1[(k0 + 1) * Hn + n];
    c = wmma_f32_16x16x4(a, b, c);
  }
  const float bn = b1[n];
#pragma unroll
  for (int i = 0; i < 8; ++i)
    u[(tm * 16 + i + 8 * hi) * Hn + n] = fmaxf(c[i] + bn, 0.f);
}

// ---------------------------------------------------------------------------
// Kernel B:  h0 = relu(u @ W2 + b2)   [T,H], K = H = 128 (32 wmma steps)
// ---------------------------------------------------------------------------
__global__ void __launch_bounds__(256) k_h_stage2(
    const float* __restrict__ u, const float* __restrict__ W2,
    const float* __restrict__ b2, float* __restrict__ h0) {
  const int wave = threadIdx.x >> 5, lane = threadIdx.x & 31;
  const int lg = lane & 15, hi = lane >> 4;
  const int tm = blockIdx.x, tn = wave;
  const int t = tm * 16 + lg, n = tn * 16 + lg;
  v8f c = {};
#pragma unroll
  for (int kk = 0; kk < Hn / 4; ++kk) {
    const int k0 = (kk << 2) + (hi << 1);
    v2f a = *(const v2f*)(u + t * Hn + k0);   // contiguous K pair -> b64 load
    v2f b;
    b.x = W2[k0 * Hn + n];
    b.y = W2[(k0 + 1) * Hn + n];
    c = wmma_f32_16x16x4(a, b, c);
  }
  const float bn = b2[n];
#pragma unroll
  for (int i = 0; i < 8; ++i)
    h0[(tm * 16 + i + 8 * hi) * Hn + n] = fmaxf(c[i] + bn, 0.f);
}

// ---------------------------------------------------------------------------
// Kernel C (the 25-GFLOP pass): per (t-block, gene-chunk) workgroup:
//   for each gene: hg = relu(z @ W2 + b2) via WMMA, score = <hg, h0>,
//   online-softmax accumulate acc/m/l per timepoint; raw scores -> ws.
// grid = (2 t-blocks of 128 rows, 250 chunks of 12 genes); 8 waves,
// each wave owns a 16-row tile x full 128 cols (8 col tiles).
// Dynamic LDS: W2 pre-paired (16384 f) + h0 block (128*132) +
//              per-wave hg buffer (8*16*132) + b1 (128)  ~= 196.5 KB.
// ---------------------------------------------------------------------------
__global__ void __launch_bounds__(256) k_flash(
    const float* __restrict__ x, const float* __restrict__ W1,
    const float* __restrict__ b1, const float* __restrict__ W2,
    const float* __restrict__ b2, const float* __restrict__ h0,
    float* __restrict__ scores, float* __restrict__ mbuf,
    float* __restrict__ lbuf, float* __restrict__ accbuf) {
  extern __shared__ float smem[];
  float* sW2p = smem;                    // 16384: pairs (W2[k0][n],W2[k0+1][n])
  float* sH0 = sW2p + Hn * Hn;           // 128 * 132
  float* sHG = sH0 + Hn * H0_LD;         // 8 waves * 16 * 132
  float* sB1 = sHG + 8 * 16 * H0_LD;     // 128

  const int tid = threadIdx.x;
  const int wave = tid >> 5, lane = tid & 31;
  const int lg = lane & 15, hi = lane >> 4;
  const int tb = blockIdx.x;      // 0..1
  const int chunk = blockIdx.y;   // 0..249

  // Stage W2 into LDS, re-packed so a B-fragment is ONE conflict-free b64.
  for (int idx = tid; idx < Hn * Hn; idx += 256) {
    const int k = idx >> 7, n = idx & 127;
    const int kk = k >> 2, r = k & 3, h = r >> 1, o = r & 1;
    sW2p[((((kk << 1) | h) * Hn) + n) * 2 + o] = W2[idx];
  }
  if (tid < Hn) sB1[tid] = b1[tid];
  for (int idx = tid; idx < 128 * Hn; idx += 256) {
    const int r = idx >> 7, n = idx & 127;
    sH0[r * H0_LD + n] = h0[(tb * 128 + r) * Hn + n];
  }
  __syncthreads();

  const int t0 = tb * 128 + wave * 16;     // first global row of this wave
  float* myHG = sHG + wave * 16 * H0_LD;
  const int rloc = wave * 16;              // row offset inside sH0 block

  float m[8], l[8];
  v8f acc[8];
#pragma unroll
  for (int i = 0; i < 8; ++i) { m[i] = -3.0e38f; l[i] = 0.f; acc[i] = (v8f){}; }

  const int g0 = chunk * GPG;
  for (int gi = 0; gi < GPG; ++gi) {
    const int g = g0 + gi;
    if (gi + 1 < GPG) {  // prefetch next gene's rows (global_prefetch_b8)
      __builtin_prefetch(x + (g + 1) * Tn + t0, 0, 1);
      __builtin_prefetch(W1 + (g + 1) * Hn, 0, 1);
    }
    const float xv = x[g * Tn + t0 + lg];

    // A fragments: z[t,k] = relu(x[g,t]*W1[g,k] + b1[k]) built in registers.
    v2f a[32];
#pragma unroll
    for (int kk = 0; kk < 32; ++kk) {
      const int k0 = (kk << 2) + (hi << 1);
      const v2f w = *(const v2f*)(W1 + g * Hn + k0);  // same addr per half
      const v2f bb = *(const v2f*)(sB1 + k0);
      a[kk].x = fmaxf(fmaf(xv, w.x, bb.x), 0.f);
      a[kk].y = fmaxf(fmaf(xv, w.y, bb.y), 0.f);
    }

    float plane[8];
#pragma unroll
    for (int i = 0; i < 8; ++i) plane[i] = 0.f;

#pragma unroll
    for (int tn = 0; tn < 8; ++tn) {
      const int n = tn * 16 + lg;
      v8f h = {};
#pragma unroll
      for (int kk = 0; kk < 32; ++kk) {
        const v2f bf2 = *(const v2f*)(sW2p + ((((kk << 1) | hi) * Hn) + n) * 2);
        h = wmma_f32_16x16x4(a[kk], bf2, h);
      }
      const float b2n = b2[n];
#pragma unroll
      for (int i = 0; i < 8; ++i) {
        const float hv = fmaxf(h[i] + b2n, 0.f);
        myHG[(i + 8 * hi) * H0_LD + n] = hv;  // spill hg tile to LDS
        plane[i] = fmaf(hv, sH0[(rloc + i + 8 * hi) * H0_LD + n], plane[i]);
      }
    }

    // Full-row dot product: butterfly over the 16-lane N group (wave32).
    float s[8], al[8], pv[8];
#pragma unroll
    for (int i = 0; i < 8; ++i) {
      float v = plane[i];
      v += __shfl_xor(v, 1);
      v += __shfl_xor(v, 2);
      v += __shfl_xor(v, 4);
      v += __shfl_xor(v, 8);
      s[i] = v;
    }
    if (lg == 0) {
#pragma unroll
      for (int i = 0; i < 8; ++i) scores[g * Tn + t0 + i + 8 * hi] = s[i];
    }
    // Online softmax state update.
#pragma unroll
    for (int i = 0; i < 8; ++i) {
      const float mn = fmaxf(m[i], s[i]);
      al[i] = __expf(m[i] - mn);
      pv[i] = __expf(s[i] - mn);
      l[i] = fmaf(l[i], al[i], pv[i]);
      m[i] = mn;
    }
    // acc = acc*alpha + p*hg  (re-read hg tiles from LDS).
#pragma unroll
    for (int tn = 0; tn < 8; ++tn) {
      const int n = tn * 16 + lg;
#pragma unroll
      for (int i = 0; i < 8; ++i) {
        const float hv = myHG[(i + 8 * hi) * H0_LD + n];
        acc[tn][i] = fmaf(acc[tn][i], al[i], pv[i] * hv);
      }
    }
  }

  // Emit split-K partials.
  if (lg == 0) {
#pragma unroll
    for (int i = 0; i < 8; ++i) {
      const int t = t0 + i + 8 * hi;
      mbuf[chunk * Tn + t] = m[i];
      lbuf[chunk * Tn + t] = l[i];
    }
  }
#pragma unroll
  for (int tn = 0; tn < 8; ++tn) {
    const int n = tn * 16 + lg;
#pragma unroll
    for (int i = 0; i < 8; ++i) {
      const int t = t0 + i + 8 * hi;
      accbuf[((size_t)chunk * Tn + t) * Hn + n] = acc[tn][i];
    }
  }
}

// ---------------------------------------------------------------------------
// Kernel D: merge split-K softmax partials; emit y_pred + global (m,l).
// One block (128 threads, thread = hidden col n) per timepoint t.
// ---------------------------------------------------------------------------
__global__ void __launch_bounds__(128) k_merge(
    const float* __restrict__ mbuf, const float* __restrict__ lbuf,
    const float* __restrict__ accbuf, const float* __restrict__ Wf,
    const float* __restrict__ bfp, float* __restrict__ y_pred,
    float* __restrict__ mglob, float* __restrict__ lglob) {
  const int t = blockIdx.x, n = threadIdx.x;
  __shared__ float red[128];
  float mloc = -3.0e38f;
  for (int c = n; c < NCHUNK; c += 128) mloc = fmaxf(mloc, mbuf[c * Tn + t]);
  red[n] = mloc;
  __syncthreads();
  for (int s2 = 64; s2 > 0; s2 >>= 1) {
    if (n < s2) red[n] = fmaxf(red[n], red[n + s2]);
    __syncthreads();
  }
  const float mg = red[0];
  __syncthreads();
  float lsum = 0.f, hn = 0.f;
  for (int c = 0; c < NCHUNK; ++c) {
    const float alpha = __expf(mbuf[c * Tn + t] - mg);    // broadcast reads
    lsum = fmaf(lbuf[c * Tn + t], alpha, lsum);
    hn = fmaf(accbuf[((size_t)c * Tn + t) * Hn + n], alpha, hn);  // coalesced
  }
  const float hatt = hn / lsum;
  red[n] = hatt * Wf[n];
  __syncthreads();
  for (int s2 = 64; s2 > 0; s2 >>= 1) {
    if (n < s2) red[n] += red[n + s2];
    __syncthreads();
  }
  if (n == 0) {
    y_pred[t] = red[0] + bfp[0];
    mglob[t] = mg;
    lglob[t] = lsum;
  }
}

// ---------------------------------------------------------------------------
// Kernel E: attention_weight[g,t] = exp(s - m[t]) / l[t]; imp_score[g] = sum_t.
// One block (256 threads = timepoints) per gene.
// ---------------------------------------------------------------------------
__global__ void __launch_bounds__(256) k_attn(
    const float* __restrict__ scores, const float* __restrict__ mglob,
    const float* __restrict__ lglob, float* __restrict__ attw,
    float* __restrict__ imp) {
  const int g = blockIdx.x, t = threadIdx.x;
  const float a = __expf(scores[g * Tn + t] - mglob[t]) / lglob[t];
  attw[g * Tn + t] = a;
  __shared__ float red[256];
  red[t] = a;
  __syncthreads();
  for (int s2 = 128; s2 > 0; s2 >>= 1) {
    if (t < s2) red[t] += red[t + s2];
    __syncthreads();
  }
  if (t == 0) imp[g] = red[0];
}

// ---------------------------------------------------------------------------
extern "C" void kernel_launch(void* const* d_in, const int* in_sizes, int n_in,
                              void* d_out, int out_size, void* d_ws,
                              size_t ws_size, hipStream_t stream) {
  (void)in_sizes; (void)n_in; (void)out_size; (void)ws_size;
  const float* x  = (const float*)d_in[0];   // [G,T]
  const float* W1 = (const float*)d_in[1];   // [G,H]
  const float* b1 = (const float*)d_in[2];   // [H]
  const float* W2 = (const float*)d_in[3];   // [H,H]
  const float* b2 = (const float*)d_in[4];   // [H]
  const float* Wf = (const float*)d_in[5];   // [H]
  const float* bf = (const float*)d_in[6];   // [1]
  float* out = (float*)d_out;                // y_pred(256) ++ attw(G*T) ++ imp(G)

  float* ws = (float*)d_ws;
  float* u      = ws;                     // 256*128
  float* h0     = u + Tn * Hn;            // 256*128
  float* scores = h0 + Tn * Hn;           // G*T = 768000
  float* mbuf   = scores + (size_t)Gn * Tn;   // NCHUNK*256
  float* lbuf   = mbuf + NCHUNK * Tn;         // NCHUNK*256
  float* mglob  = lbuf + NCHUNK * Tn;         // 256
  float* lglob  = mglob + Tn;                 // 256
  float* accbuf = lglob + Tn;                 // NCHUNK*256*128 (~32.8 MB)

  k_h_stage1<<<dim3(Tn / 16), 256, 0, stream>>>(x, W1, b1, u);
  k_h_stage2<<<dim3(Tn / 16), 256, 0, stream>>>(u, W2, b2, h0);

  const size_t ldsC =
      (size_t)(Hn * Hn + Hn * H0_LD + 8 * 16 * H0_LD + Hn) * sizeof(float);
  k_flash<<<dim3(2, NCHUNK), 256, ldsC, stream>>>(x, W1, b1, W2, b2, h0, scores,
                                                  mbuf, lbuf, accbuf);

  k_merge<<<dim3(Tn), 128, 0, stream>>>(mbuf, lbuf, accbuf, Wf, bf, out, mglob,
                                        lglob);
  k_attn<<<dim3(Gn), 256, 0, stream>>>(scores, mglob, lglob, out + Tn,
                                       out + Tn + (size_t)Gn * Tn);
}